// LuongAttention_10565619548604
// MI455X (gfx1250) — compile-verified
//
#include <hip/hip_runtime.h>
#include <hip/hip_bf16.h>

typedef float v2f __attribute__((ext_vector_type(2)));
typedef float v8f __attribute__((ext_vector_type(8)));

#define NROWS   131072
#define HDIM    512
#define NSEG    64
#define APITCH  516   // 16 rows * 516 floats: lane m hits banks (4m,4m+1) -> conflict-free b64 reads

// ---------------------------------------------------------------------------
// Kernel 1: transpose W (512 x 1024) into
//   W1t[k*512 + j]                      = W[j*1024 + k]        (k < 512)  for bias GEMM
//   W2p[(k>>1)*1024 + j*2 + (k&1)]      = W[j*1024 + 512 + k]  (K-paired for WMMA B frags)
// ---------------------------------------------------------------------------
__global__ __launch_bounds__(256) void luong_transpose(const float* __restrict__ W,
                                                       float* __restrict__ W1t,
                                                       float* __restrict__ W2p) {
    __shared__ float tile[32][33];
    const int c0 = blockIdx.x * 32;   // column of W (0..1023)
    const int j0 = blockIdx.y * 32;   // row of W    (0..511)
    const int tx = threadIdx.x;       // 0..31
    const int ty = threadIdx.y;       // 0..7
    for (int yy = ty; yy < 32; yy += 8)
        tile[yy][tx] = W[(size_t)(j0 + yy) * 1024 + (c0 + tx)];
    __syncthreads();
    for (int yy = ty; yy < 32; yy += 8) {
        const int c = c0 + yy;
        const int j = j0 + tx;
        const float val = tile[tx][yy];          // = W[j][c]
        if (c < 512) {
            W1t[c * 512 + j] = val;
        } else {
            const int k = c - 512;
            W2p[(k >> 1) * 1024 + j * 2 + (k & 1)] = val;
        }
    }
}

// ---------------------------------------------------------------------------
// Kernel 2: bias[b][j] = sum_k hid[b][k] * W[j][k]   (64 x 512, tiny)
// ---------------------------------------------------------------------------
__global__ __launch_bounds__(256) void luong_bias(const float* __restrict__ hid,
                                                  const float* __restrict__ W1t,
                                                  float* __restrict__ bias) {
    const int idx = blockIdx.x * 256 + threadIdx.x;   // 0..32767
    const int b = idx >> 9;
    const int j = idx & 511;
    float s = 0.f;
    #pragma unroll 8
    for (int k = 0; k < 512; ++k)
        s = fmaf(hid[b * 512 + k], W1t[k * 512 + j], s);
    bias[idx] = s;
}

// ---------------------------------------------------------------------------
// Kernel 3 (main): per 16-row block, scores[i] = sum_j tanh(enc[i]@W2.T + bias[seg,j]) * v[j]
// f32 WMMA 16x16x4; 8 waves, each wave owns 4 j-tiles (64 columns).
// ---------------------------------------------------------------------------
__global__ __launch_bounds__(256) void luong_scores(const float* __restrict__ enc,
                                                    const int*   __restrict__ seg,
                                                    const float* __restrict__ W2p,
                                                    const float* __restrict__ bias,
                                                    const float* __restrict__ v,
                                                    float*       __restrict__ scores) {
    __shared__ float sA[16 * APITCH];
    __shared__ float sScore[16];
    __shared__ int   sSeg[16];

    const int tid  = threadIdx.x;
    const int row0 = blockIdx.x * 16;

    if (tid < 16) {
        sScore[tid] = 0.f;
        sSeg[tid]   = seg[row0 + tid];
    }
    // Stage 16x512 enc tile into LDS (float4, coalesced)
    {
        const float4* src = (const float4*)(enc + (size_t)row0 * HDIM);
        for (int l = tid; l < 16 * 128; l += 256) {
            const int r  = l >> 7;
            const int c4 = l & 127;
            const float4 val = src[r * 128 + c4];
            *(float4*)&sA[r * APITCH + c4 * 4] = val;
        }
    }
    __syncthreads();

    const int wave = tid >> 5;
    const int lane = tid & 31;
    const int nn   = lane & 15;   // column within tile / A-row for A frag
    const int hi   = lane >> 4;   // K-half selector (ISA f32 A/B frag layout)

    const float* aptr = sA + nn * APITCH + 2 * hi;          // A[m=nn][k0 + 2*hi + {0,1}]
    float sacc[8];
    #pragma unroll
    for (int r = 0; r < 8; ++r) sacc[r] = 0.f;

    #pragma unroll
    for (int t = 0; t < 4; ++t) {
        const int jt = wave * 64 + t * 16;
        const float* bptr = W2p + (size_t)(jt + nn) * 2 + hi * 1024;  // B[k0+2*hi+{0,1}][n=nn]
        v8f c = {};
        #pragma unroll 4
        for (int k0 = 0; k0 < HDIM; k0 += 4) {
            v2f a = *(const v2f*)(aptr + k0);
            v2f b = *(const v2f*)(bptr + (k0 >> 1) * 1024);
            c = __builtin_amdgcn_wmma_f32_16x16x4_f32(false, a, false, b,
                                                      (short)0, c, false, false);
        }
        // C layout: element c[r] is (row = r + 8*hi, col = jt + nn)
        const float vj = v[jt + nn];
        #pragma unroll
        for (int r = 0; r < 8; ++r) {
            const int mrow = r + 8 * hi;
            const float bz = bias[sSeg[mrow] * HDIM + jt + nn];
            sacc[r] += tanhf(c[r] + bz) * vj;
        }
    }

    // Reduce over the 16 lanes sharing each row half (xor masks stay inside halves)
    #pragma unroll
    for (int r = 0; r < 8; ++r) {
        float x = sacc[r];
        x += __shfl_xor(x, 1, 32);
        x += __shfl_xor(x, 2, 32);
        x += __shfl_xor(x, 4, 32);
        x += __shfl_xor(x, 8, 32);
        if (nn == 0) atomicAdd(&sScore[8 * hi + r], x);
    }
    __syncthreads();
    if (tid < 16) scores[row0 + tid] = sScore[tid];
}

// ---------------------------------------------------------------------------
// Segmented softmax (ordered-uint max trick, two-level atomics)
// ---------------------------------------------------------------------------
__device__ __forceinline__ unsigned enc_ord(float f) {
    unsigned u = __float_as_uint(f);
    return (u & 0x80000000u) ? ~u : (u | 0x80000000u);
}
__device__ __forceinline__ float dec_ord(unsigned u) {
    unsigned b = (u & 0x80000000u) ? (u & 0x7FFFFFFFu) : ~u;
    return __uint_as_float(b);
}

__global__ void luong_stats_init(unsigned* gmax, float* gden) {
    const int t = threadIdx.x;
    if (t < NSEG) { gmax[t] = 0u; gden[t] = 0.f; }
}

__global__ __launch_bounds__(256) void luong_segmax(const float* __restrict__ scores,
                                                    const int*   __restrict__ seg,
                                                    unsigned*    __restrict__ gmax) {
    __shared__ unsigned sm[NSEG];
    const int tid = threadIdx.x;
    if (tid < NSEG) sm[tid] = 0u;
    __syncthreads();
    const int i = blockIdx.x * 256 + tid;
    atomicMax(&sm[seg[i]], enc_ord(scores[i]));
    __syncthreads();
    if (tid < NSEG && sm[tid] != 0u) atomicMax(&gmax[tid], sm[tid]);
}

__global__ __launch_bounds__(256) void luong_expsum(const float*    __restrict__ scores,
                                                    const int*      __restrict__ seg,
                                                    const unsigned* __restrict__ gmax,
                                                    float*          __restrict__ e_out,
                                                    float*          __restrict__ gden) {
    __shared__ float ss[NSEG];
    const int tid = threadIdx.x;
    if (tid < NSEG) ss[tid] = 0.f;
    __syncthreads();
    const int i = blockIdx.x * 256 + tid;
    const int s = seg[i];
    const float e = expf(scores[i] - dec_ord(gmax[s]));
    e_out[i] = e;
    atomicAdd(&ss[s], e);
    __syncthreads();
    if (tid < NSEG && ss[tid] != 0.f) atomicAdd(&gden[tid], ss[tid]);
}

__global__ __launch_bounds__(256) void luong_norm(const int*   __restrict__ seg,
                                                  const float* __restrict__ gden,
                                                  float*       __restrict__ out) {
    const int i = blockIdx.x * 256 + threadIdx.x;
    out[i] = out[i] / gden[seg[i]];
}

// ---------------------------------------------------------------------------
extern "C" void kernel_launch(void* const* d_in, const int* in_sizes, int n_in,
                              void* d_out, int out_size, void* d_ws, size_t ws_size,
                              hipStream_t stream) {
    const float* hid = (const float*)d_in[0];   // (64, 512)
    const float* enc = (const float*)d_in[1];   // (131072, 512)
    const int*   seg = (const int*)  d_in[2];   // (131072,)
    const float* W   = (const float*)d_in[3];   // (512, 1024)
    const float* v   = (const float*)d_in[4];   // (512, 1)
    float* out = (float*)d_out;                 // (131072, 1)

    float* ws     = (float*)d_ws;
    float* W1t    = ws;                         // 512*512
    float* W2p    = W1t + 512 * 512;            // 512*512
    float* biasT  = W2p + 512 * 512;            // 64*512
    float* scores = biasT + 64 * 512;           // 131072
    unsigned* gmax = (unsigned*)(scores + NROWS);   // 64
    float*    gden = (float*)(gmax + NSEG);         // 64

    luong_transpose<<<dim3(32, 16), dim3(32, 8), 0, stream>>>(W, W1t, W2p);
    luong_bias<<<128, 256, 0, stream>>>(hid, W1t, biasT);
    luong_scores<<<NROWS / 16, 256, 0, stream>>>(enc, seg, W2p, biasT, v, scores);
    luong_stats_init<<<1, 64, 0, stream>>>(gmax, gden);
    luong_segmax<<<NROWS / 256, 256, 0, stream>>>(scores, seg, gmax);
    luong_expsum<<<NROWS / 256, 256, 0, stream>>>(scores, seg, gmax, out, gden);
    luong_norm<<<NROWS / 256, 256, 0, stream>>>(seg, gden, out);
}